// DGNN_16200616641060
// MI455X (gfx1250) — compile-verified
//
#include <hip/hip_runtime.h>
#include <stdint.h>

typedef float v2f __attribute__((ext_vector_type(2)));
typedef float v8f __attribute__((ext_vector_type(8)));

#define WMMA_F32(a, b, c) \
  __builtin_amdgcn_wmma_f32_16x16x4_f32(false, (a), false, (b), (short)0, (c), false, false)

// ---------------- utility kernels ----------------

__global__ void k_fill_f32(float* __restrict__ p, float v, int n) {
  int i = blockIdx.x * blockDim.x + threadIdx.x;
  if (i < n) p[i] = v;
}

__global__ void k_fill_i32(int* __restrict__ p, int v, int n) {
  int i = blockIdx.x * blockDim.x + threadIdx.x;
  if (i < n) p[i] = v;
}

// degree on destination nodes
__global__ void k_deg(const int* __restrict__ dst, float* __restrict__ deg, int E) {
  int e = blockIdx.x * blockDim.x + threadIdx.x;
  if (e < E) atomicAdd(&deg[dst[e]], 1.0f);
}

// dinv = rsqrt(deg + 1)  (in place)
__global__ void k_dinv(float* __restrict__ d, int n) {
  int i = blockIdx.x * blockDim.x + threadIdx.x;
  if (i < n) d[i] = rsqrtf(d[i] + 1.0f);
}

// starts[b] = min node index with batch == b (batch is sorted contiguous)
__global__ void k_starts(const int* __restrict__ batch, int* __restrict__ starts, int n) {
  int i = blockIdx.x * blockDim.x + threadIdx.x;
  if (i < n) atomicMin(&starts[batch[i]], i);
}

// pad x [N,F] -> xp [N,Fp] (zeros in padded cols)
__global__ void k_pad_x(const float* __restrict__ x, float* __restrict__ xp,
                        int N, int F, int Fp) {
  int i = blockIdx.x * blockDim.x + threadIdx.x;
  if (i < N * Fp) {
    int n = i / Fp, f = i % Fp;
    xp[i] = (f < F) ? x[n * F + f] : 0.0f;
  }
}

// Pack W [Fin,Fout] into WMMA B-fragment order, zero padded to [Kp,Np]:
//   out[((c*T + t)*32 + lane)*2 + p] = W[4c + (lane>>4)*2 + p][16t + (lane&15)]
// (c = 4-row k-chunk, t = 16-col tile, T = Np/16). One contiguous b64 per lane.
__global__ void k_pack_w(const float* __restrict__ w, float* __restrict__ wp,
                         int Fin, int Fout, int Kp, int Np) {
  int i = blockIdx.x * blockDim.x + threadIdx.x;
  if (i >= Kp * Np) return;
  int p    = i & 1;
  int lane = (i >> 1) & 31;
  int T    = Np >> 4;
  int t    = (i >> 6) % T;
  int c    = (i >> 6) / T;
  int k = 4 * c + ((lane >> 4) * 2) + p;
  int n = 16 * t + (lane & 15);
  wp[i] = (k < Fin && n < Fout) ? w[k * Fout + n] : 0.0f;
}

// ---------------- shared WMMA strip accumulator --------------------------------
// Double buffered + pointer-bumped: loads for chunk c+1 issue before chunk c's
// WMMAs; strides are loop-invariant so unrolled iterations get immediate offsets.
template <int STRIP>
__device__ __forceinline__ void gemm_strip(const float* __restrict__ arow,
                                           const float* __restrict__ wrow,
                                           int chunks, int wstride,
                                           v8f acc[STRIP]) {
  v2f a_cur = *(const v2f*)arow;
  v2f b_cur[STRIP];
  #pragma unroll
  for (int t = 0; t < STRIP; t++) b_cur[t] = *(const v2f*)(wrow + t * 64);

  for (int c = 0; c < chunks - 1; ++c) {
    arow += 4;
    wrow += wstride;
    v2f a_nxt = *(const v2f*)arow;
    v2f b_nxt[STRIP];
    #pragma unroll
    for (int t = 0; t < STRIP; t++) b_nxt[t] = *(const v2f*)(wrow + t * 64);
    #pragma unroll
    for (int t = 0; t < STRIP; t++) acc[t] = WMMA_F32(a_cur, b_cur[t], acc[t]);
    a_cur = a_nxt;
    #pragma unroll
    for (int t = 0; t < STRIP; t++) b_cur[t] = b_nxt[t];
  }
  #pragma unroll
  for (int t = 0; t < STRIP; t++) acc[t] = WMMA_F32(a_cur, b_cur[t], acc[t]);
}

// ---------------- WMMA fp32 GEMM: C[N,Np] = A[N,Kp] * Wf(packed) ----------------
// 128 threads = 4 waves; each wave owns a 16 x (16*STRIP) output strip.
template <int STRIP>
__global__ void k_gemm(const float* __restrict__ A, const float* __restrict__ Wf,
                       float* __restrict__ C, int Nrows, int Kp, int Np) {
  const int lane = threadIdx.x & 31;
  const int wave = threadIdx.x >> 5;
  const int m0 = (blockIdx.x * 4 + wave) * 16;
  if (m0 >= Nrows) return;                        // wave-uniform
  const int t0 = blockIdx.y * STRIP;              // first 16-col tile
  const int ml = lane & 15;
  const int kb = (lane >> 4) * 2;

  v8f acc[STRIP];
  #pragma unroll
  for (int t = 0; t < STRIP; t++) acc[t] = (v8f){};

  gemm_strip<STRIP>(A + (size_t)(m0 + ml) * Kp + kb,
                    Wf + ((size_t)t0 * 64) + lane * 2,
                    Kp >> 2, (Np >> 4) * 64, acc);

  const int rbase = m0 + ((lane >> 4) * 8);
  #pragma unroll
  for (int t = 0; t < STRIP; t++) {
    const int col = (t0 + t) * 16 + ml;
    #pragma unroll
    for (int i = 0; i < 8; i++)
      C[(size_t)(rbase + i) * Np + col] = acc[t][i];
  }
}

// ---------------- edge aggregation: out[dst] += pre[src] * dinv[src]*dinv[dst] --------
__global__ void k_agg(const float* __restrict__ pre, float* __restrict__ out,
                      const int* __restrict__ src, const int* __restrict__ dst,
                      const float* __restrict__ dinv, int E, int Fout, int Fp) {
  int e = blockIdx.x * 8 + (threadIdx.x >> 5);
  if (e >= E) return;
  int lane = threadIdx.x & 31;
  int s = src[e], d = dst[e];
  float coef = dinv[s] * dinv[d];
  const float* __restrict__ prow = pre + (size_t)s * Fp;
  float* __restrict__ orow = out + (size_t)d * Fp;
  for (int f = lane; f < Fout; f += 32)
    atomicAdd(&orow[f], prow[f] * coef);
}

// ---------------- finalize: out = relu(agg + pre*dinv^2 + bias) (in place) ----------
__global__ void k_fin(float* __restrict__ agg, const float* __restrict__ pre,
                      const float* __restrict__ dinv, const float* __restrict__ bias,
                      int N, int Fout, int Fp, int relu) {
  int i = blockIdx.x * blockDim.x + threadIdx.x;
  if (i >= N * Fout) return;
  int n = i / Fout, f = i % Fout;
  size_t idx = (size_t)n * Fp + f;
  float di = dinv[n];
  float v = agg[idx] + pre[idx] * di * di + bias[f];
  if (relu) v = fmaxf(v, 0.0f);
  agg[idx] = v;
}

// ---------------- fused FC + to_dense_batch scatter (STRIP=5 strips) ---------------
__global__ void k_fc(const float* __restrict__ A, const float* __restrict__ Wf,
                     const float* __restrict__ bias, const int* __restrict__ batch,
                     const int* __restrict__ starts, const int* __restrict__ pmaxnum,
                     float* __restrict__ out, int Nrows, int Kp, int Npad, int OUTd) {
  constexpr int STRIP = 5;
  const int lane = threadIdx.x & 31;
  const int wave = threadIdx.x >> 5;
  const int m0 = (blockIdx.x * 4 + wave) * 16;
  if (m0 >= Nrows) return;                        // wave-uniform
  const int t0 = blockIdx.y * STRIP;
  const int ml = lane & 15;
  const int kb = (lane >> 4) * 2;

  v8f acc[STRIP];
  #pragma unroll
  for (int t = 0; t < STRIP; t++) acc[t] = (v8f){};

  gemm_strip<STRIP>(A + (size_t)(m0 + ml) * Kp + kb,
                    Wf + ((size_t)t0 * 64) + lane * 2,
                    Kp >> 2, (Npad >> 4) * 64, acc);

  const int mn = *pmaxnum;
  const int rbase = m0 + ((lane >> 4) * 8);
  #pragma unroll
  for (int t = 0; t < STRIP; t++) {
    const int col = (t0 + t) * 16 + ml;
    if (col < OUTd) {
      float bi = bias[col];
      #pragma unroll
      for (int i = 0; i < 8; i++) {
        int row = rbase + i;
        int b = batch[row];
        int pos = row - starts[b];
        if (pos < mn)
          out[((size_t)b * mn + pos) * OUTd + col] = acc[t][i] + bi;
      }
    }
  }
}

// ---------------- host launcher ----------------

extern "C" void kernel_launch(void* const* d_in, const int* in_sizes, int n_in,
                              void* d_out, int out_size, void* d_ws, size_t ws_size,
                              hipStream_t stream) {
  const float* x    = (const float*)d_in[0];
  const int*   ei   = (const int*)  d_in[1];
  const int*   batch= (const int*)  d_in[2];
  const int*   pmax = (const int*)  d_in[3];
  const float* W1   = (const float*)d_in[4];  const float* b1  = (const float*)d_in[5];
  const float* W2   = (const float*)d_in[6];  const float* b2  = (const float*)d_in[7];
  const float* W3   = (const float*)d_in[8];  const float* b3  = (const float*)d_in[9];
  const float* Wfc  = (const float*)d_in[10]; const float* bfc = (const float*)d_in[11];

  const int F0   = in_sizes[5];            // 78
  const int F2   = in_sizes[7];            // 156
  const int F3   = in_sizes[9];            // 312
  const int OUTd = in_sizes[11];           // 200
  const int N    = in_sizes[0] / F0;       // 102400
  const int E    = in_sizes[1] / 2;        // 409600
  const int* src = ei;
  const int* dst = ei + E;

  auto pad16 = [](int v) { return (v + 15) & ~15; };
  auto pad80 = [](int v) { return ((v + 79) / 80) * 80; };
  const int F0p  = pad16(F0);              // 80  = 1 strip of 5 tiles
  const int F2p  = pad16(F2);              // 160 = 2 strips
  const int F3p  = pad16(F3);              // 320 = 4 strips
  const int OUTp = pad80(OUTd);            // 240 = 3 strips (cols >= 200 masked)

  // workspace layout (floats)
  float* ws = (float*)d_ws;
  size_t o = 0;
  float* dinv   = ws + o; o += (size_t)N;
  int*   starts = (int*)(ws + o); o += (size_t)N;
  float* W1p = ws + o; o += (size_t)F0p * F0p;
  float* W2p = ws + o; o += (size_t)F0p * F2p;
  float* W3p = ws + o; o += (size_t)F2p * F3p;
  float* Wfp = ws + o; o += (size_t)F3p * OUTp;
  float* SA  = ws + o; o += (size_t)N * F3p;   // xpad(80) -> pre2(160) -> h3(320)
  float* SB  = ws + o; o += (size_t)N * F2p;   // pre1(80) -> h2(160)
  float* SC  = ws + o; o += (size_t)N * F3p;   // h1(80)   -> pre3(320)
  (void)ws_size;

  const dim3 blk(256);
  auto g1 = [](int n) { return dim3((unsigned)((n + 255) / 256)); };
  const dim3 gblk(128);
  const int rowBlocks = ((N / 16) + 3) / 4;
  auto ggemm = [&](int Np) { return dim3((unsigned)rowBlocks, (unsigned)(Np / 80)); };

  // degrees -> dinv
  k_fill_f32<<<g1(N), blk, 0, stream>>>(dinv, 0.0f, N);
  k_deg     <<<g1(E), blk, 0, stream>>>(dst, dinv, E);
  k_dinv    <<<g1(N), blk, 0, stream>>>(dinv, N);

  // per-graph start indices
  k_fill_i32<<<g1(N), blk, 0, stream>>>(starts, 0x7fffffff, N);
  k_starts  <<<g1(N), blk, 0, stream>>>(batch, starts, N);

  // pad inputs / pack weights into WMMA fragment order
  k_pad_x <<<g1(N * F0p), blk, 0, stream>>>(x, SA, N, F0, F0p);
  k_pack_w<<<g1(F0p * F0p), blk, 0, stream>>>(W1, W1p, F0, F0, F0p, F0p);
  k_pack_w<<<g1(F0p * F2p), blk, 0, stream>>>(W2, W2p, F0, F2, F0p, F2p);
  k_pack_w<<<g1(F2p * F3p), blk, 0, stream>>>(W3, W3p, F2, F3, F2p, F3p);
  k_pack_w<<<g1(F3p * OUTp), blk, 0, stream>>>(Wfc, Wfp, F3, OUTd, F3p, OUTp);

  // dense output must be zero in padding slots
  k_fill_f32<<<g1(out_size), blk, 0, stream>>>((float*)d_out, 0.0f, out_size);

  // ---- layer 1: SA(xpad) -> SB(pre1) -> SC(h1) ----
  k_gemm<5><<<ggemm(F0p), gblk, 0, stream>>>(SA, W1p, SB, N, F0p, F0p);
  k_fill_f32<<<g1(N * F0p), blk, 0, stream>>>(SC, 0.0f, N * F0p);
  k_agg<<<dim3((unsigned)((E + 7) / 8)), blk, 0, stream>>>(SB, SC, src, dst, dinv, E, F0, F0p);
  k_fin<<<g1(N * F0), blk, 0, stream>>>(SC, SB, dinv, b1, N, F0, F0p, 1);

  // ---- layer 2: SC(h1) -> SA(pre2) -> SB(h2) ----
  k_gemm<5><<<ggemm(F2p), gblk, 0, stream>>>(SC, W2p, SA, N, F0p, F2p);
  k_fill_f32<<<g1(N * F2p), blk, 0, stream>>>(SB, 0.0f, N * F2p);
  k_agg<<<dim3((unsigned)((E + 7) / 8)), blk, 0, stream>>>(SA, SB, src, dst, dinv, E, F2, F2p);
  k_fin<<<g1(N * F2), blk, 0, stream>>>(SB, SA, dinv, b2, N, F2, F2p, 1);

  // ---- layer 3: SB(h2) -> SC(pre3) -> SA(h3) ----
  k_gemm<5><<<ggemm(F3p), gblk, 0, stream>>>(SB, W3p, SC, N, F2p, F3p);
  k_fill_f32<<<g1(N * F3p), blk, 0, stream>>>(SA, 0.0f, N * F3p);
  k_agg<<<dim3((unsigned)((E + 7) / 8)), blk, 0, stream>>>(SC, SA, src, dst, dinv, E, F3, F3p);
  k_fin<<<g1(N * F3), blk, 0, stream>>>(SA, SC, dinv, b3, N, F3, F3p, 1);

  // ---- FC + scatter into dense batch ----
  k_fc<<<dim3((unsigned)rowBlocks, (unsigned)(OUTp / 80)), gblk, 0, stream>>>(
      SA, Wfp, bfc, batch, starts, pmax, (float*)d_out, N, F3p, OUTp, OUTd);
}